// PITLoss_77919296684577
// MI455X (gfx1250) — compile-verified
//
#include <hip/hip_runtime.h>

// ---------------------------------------------------------------------------
// PIT (permutation-invariant) BCE loss, B=64, T=8000, K=4, fp32.
//
// cost[b][i][j] = sum_t m * ( L_i * (-(1+0.5*t_j)) + x_i * (1 - t_j) ),
// L = log_sigmoid(x).  This is a (K x T)*(T x K) GEMM per batch, mapped onto
// V_WMMA_F32_16X16X4_F32 with the 16x16 tile packed as
//   rows/cols 0-3: L / -m(1+0.5t)   (batch b1)
//   rows/cols 4-7: x /  m(1-t)      (batch b1)
//   rows/cols 8-15: same for batch b2
// cost_b[i][j] = D[i][j] + D[4+i][4+j] (diagonal blocks; cross blocks unused).
//
// Data path: coalesced float4 global loads -> per-element transforms once ->
// LDS (bank-conflict-padded) -> ds_load_2addr pairs -> WMMA.  Partials per
// (pair, slice) written to exclusive ws slots -> deterministic finalize.
// ---------------------------------------------------------------------------

typedef __attribute__((ext_vector_type(2))) float v2f;
typedef __attribute__((ext_vector_type(4))) float v4f;
typedef __attribute__((ext_vector_type(8))) float v8f;

#define B_DIM    64
#define T_DIM    8000
#define S_SLICES 50        // 8000 / 50 = 160 timesteps per slice
#define T_SLICE  (T_DIM / S_SLICES)
#define TT       32        // timesteps staged per LDS round (5 rounds/slice)

// LDS layout: 4 arrays (0=L, 1=x, 2=bN, 3=bP) x 2 batches x 32 t x 4 ch.
// batch stride 132 (=128+4), array stride 284 (=2*132+20): distinct bank
// offsets for the 4-lane read groups of the A-set and of the B-set.
#define ARR_STRIDE 284
#define BAT_STRIDE 132
#define LIDX(arr, b) ((arr) * ARR_STRIDE + (b) * BAT_STRIDE)
#define LDS_FLOATS (3 * ARR_STRIDE + BAT_STRIDE + TT * 4)   // 1112 -> pad 1136

__device__ __forceinline__ float log_sigmoid_f(float x) {
    // log_sigmoid(x) = min(x,0) - log1p(exp(-|x|))   (stable)
    float ax = fabsf(x);
    return fminf(x, 0.0f) - __logf(1.0f + __expf(-ax));
}

__device__ __forceinline__ v4f log_sigmoid4(v4f x) {
    v4f r;
    r.x = log_sigmoid_f(x.x);
    r.y = log_sigmoid_f(x.y);
    r.z = log_sigmoid_f(x.z);
    r.w = log_sigmoid_f(x.w);
    return r;
}

__global__ void __launch_bounds__(32)
pit_cost_wmma(const float* __restrict__ logits,
              const float* __restrict__ targets,
              const float* __restrict__ vmask,
              float* __restrict__ costPart,    // [S_SLICES][B_DIM][16]
              float* __restrict__ denomPart)   // [S_SLICES][B_DIM]
{
    __shared__ float smem[1136];

    const int lane  = threadIdx.x;        // 0..31 (wave32, one wave per block)
    const int pair  = blockIdx.x;         // 0..31 -> batches (2p, 2p+1)
    const int slice = blockIdx.y;         // 0..S_SLICES-1
    const int b1 = 2 * pair, b2 = b1 + 1;
    const int tBeg = slice * T_SLICE;

    // A-matrix 16x4 f32 layout: lanes 0-15 -> M=lane, v0=K0,v1=K1;
    // lanes 16-31 -> M=lane-16, v0=K2,v1=K3.  B (4x16) mirrors transposed.
    const int half = lane >> 4;           // selects K pair {0,1} vs {2,3}
    const int mn   = lane & 15;           // A row == B column for this lane

    // A-fragment role: rows 0-3 = L_i(b1), 4-7 = x_i(b1), 8-15 = same for b2.
    const int  loc  = (mn < 8) ? 0 : 1;   // batch within the pair
    const int  sub  = mn & 7;
    const bool isL  = (sub < 4);          // also "isNeg" for the B column
    const int  ch   = sub & 3;

    // Per-lane LDS read bases (element at k:  base + k*4, pair partner +4)
    const int paBase = LIDX(isL ? 0 : 1, loc) + ch + 8 * half;
    const int pbBase = LIDX(isL ? 2 : 3, loc) + ch + 8 * half;

    const float* __restrict__ lg1 = logits  + (size_t)b1 * (T_DIM * 4);
    const float* __restrict__ lg2 = logits  + (size_t)b2 * (T_DIM * 4);
    const float* __restrict__ tg1 = targets + (size_t)b1 * (T_DIM * 4);
    const float* __restrict__ tg2 = targets + (size_t)b2 * (T_DIM * 4);
    const float* __restrict__ mk1 = vmask   + (size_t)b1 * T_DIM;
    const float* __restrict__ mk2 = vmask   + (size_t)b2 * T_DIM;

    v8f acc = {0.f, 0.f, 0.f, 0.f, 0.f, 0.f, 0.f, 0.f};
    float sumM1 = 0.f, sumM2 = 0.f;

    for (int tb = tBeg; tb < tBeg + T_SLICE; tb += TT) {
        const int t = tb + lane;          // this lane stages timestep t

        // ---- coalesced global loads: one timestep's 4 channels per lane ----
        v4f x1 = *(const v4f*)(lg1 + (size_t)t * 4);
        v4f x2 = *(const v4f*)(lg2 + (size_t)t * 4);
        v4f t1 = *(const v4f*)(tg1 + (size_t)t * 4);
        v4f t2 = *(const v4f*)(tg2 + (size_t)t * 4);
        float m1 = mk1[t];
        float m2 = mk2[t];
        sumM1 += m1;
        sumM2 += m2;

        // prefetch next stage while we compute (gfx1250 global_prefetch_b8)
        if (tb + TT < tBeg + T_SLICE) {
            __builtin_prefetch(lg1 + (size_t)(t + TT) * 4, 0, 3);
            __builtin_prefetch(lg2 + (size_t)(t + TT) * 4, 0, 3);
            __builtin_prefetch(tg1 + (size_t)(t + TT) * 4, 0, 3);
            __builtin_prefetch(tg2 + (size_t)(t + TT) * 4, 0, 3);
        }

        // ---- per-element transforms, exactly once ----
        v4f L1 = log_sigmoid4(x1);
        v4f L2 = log_sigmoid4(x2);
        v4f bn1 = -m1 * (1.0f + 0.5f * t1);
        v4f bn2 = -m2 * (1.0f + 0.5f * t2);
        v4f bp1 =  m1 * (1.0f - t1);
        v4f bp2 =  m2 * (1.0f - t2);

        // ---- stage into LDS (aligned b128 stores) ----
        const int l4 = lane * 4;
        *(v4f*)&smem[LIDX(0, 0) + l4] = L1;
        *(v4f*)&smem[LIDX(0, 1) + l4] = L2;
        *(v4f*)&smem[LIDX(1, 0) + l4] = x1;
        *(v4f*)&smem[LIDX(1, 1) + l4] = x2;
        *(v4f*)&smem[LIDX(2, 0) + l4] = bn1;
        *(v4f*)&smem[LIDX(2, 1) + l4] = bn2;
        *(v4f*)&smem[LIDX(3, 0) + l4] = bp1;
        *(v4f*)&smem[LIDX(3, 1) + l4] = bp2;
        // single wave per workgroup: DScnt dependency wait is automatic

        // ---- 8 x WMMA over the 32 staged timesteps (K=4 each) ----
#pragma unroll
        for (int kk = 0; kk < TT; kk += 4) {
            v2f av, bv;
            av[0] = smem[paBase + kk * 4];
            av[1] = smem[paBase + kk * 4 + 4];
            bv[0] = smem[pbBase + kk * 4];
            bv[1] = smem[pbBase + kk * 4 + 4];
            acc = __builtin_amdgcn_wmma_f32_16x16x4_f32(
                false, av, false, bv, (short)0, acc, false, false);
        }
    }

    // D layout: VGPR r, lanes 0-15 -> M=r, N=lane; lanes 16-31 -> M=r+8, N=lane-16.
    // cost_b1[i][j] = D[i][j]   (VGPR i,  lane j)    + D[4+i][4+j]   (VGPR 4+i, lane j+4)
    // cost_b2[i][j] = D[8+i][8+j](VGPR i, lane 24+j) + D[12+i][12+j] (VGPR 4+i, lane 28+j)
    float comb[4];
#pragma unroll
    for (int i = 0; i < 4; ++i) {
        float xpart = __shfl(acc[4 + i], lane + 4, 32);
        comb[i] = acc[i] + xpart;
    }

    if (lane < 4) {
        float* dst = costPart + ((size_t)slice * B_DIM + b1) * 16 + lane;
#pragma unroll
        for (int i = 0; i < 4; ++i) dst[i * 4] = comb[i];
    } else if (lane >= 24 && lane < 28) {
        const int j = lane - 24;
        float* dst = costPart + ((size_t)slice * B_DIM + b2) * 16 + j;
#pragma unroll
        for (int i = 0; i < 4; ++i) dst[i * 4] = comb[i];
    }

    // ---- per-slice mask sums (denominator partials) ----
#pragma unroll
    for (int off = 16; off > 0; off >>= 1) {
        sumM1 += __shfl_down(sumM1, off, 32);
        sumM2 += __shfl_down(sumM2, off, 32);
    }
    if (lane == 0) {
        denomPart[(size_t)slice * B_DIM + b1] = sumM1;
        denomPart[(size_t)slice * B_DIM + b2] = sumM2;
    }
}

__global__ void __launch_bounds__(64)
pit_finalize(const float* __restrict__ costPart,
             const float* __restrict__ denomPart,
             float* __restrict__ out)
{
    __shared__ float red[B_DIM];
    const int b = threadIdx.x;   // one thread per batch

    float c[16];
#pragma unroll
    for (int e = 0; e < 16; ++e) c[e] = 0.f;
    float d = 0.f;

    for (int s = 0; s < S_SLICES; ++s) {
        const float* src = costPart + ((size_t)s * B_DIM + b) * 16;
#pragma unroll
        for (int e = 0; e < 16; ++e) c[e] += src[e];
        d += denomPart[(size_t)s * B_DIM + b];
    }

    const float inv = 1.0f / fmaxf(d, 1.0f);
#pragma unroll
    for (int e = 0; e < 16; ++e) c[e] *= inv;

    // min over the 24 permutations of {0,1,2,3}
    float best = 3.4e38f;
    for (int a0 = 0; a0 < 4; ++a0)
        for (int a1 = 0; a1 < 4; ++a1) {
            if (a1 == a0) continue;
            for (int a2 = 0; a2 < 4; ++a2) {
                if (a2 == a0 || a2 == a1) continue;
                const int a3 = 6 - a0 - a1 - a2;
                float s = c[0 * 4 + a0] + c[1 * 4 + a1] +
                          c[2 * 4 + a2] + c[3 * 4 + a3];
                best = fminf(best, s * 0.25f);
            }
        }

    red[b] = best;
    __syncthreads();
    if (b == 0) {
        float s = 0.f;
        for (int i = 0; i < B_DIM; ++i) s += red[i];
        out[0] = s / (float)B_DIM;
    }
}

extern "C" void kernel_launch(void* const* d_in, const int* in_sizes, int n_in,
                              void* d_out, int out_size, void* d_ws, size_t ws_size,
                              hipStream_t stream) {
    (void)in_sizes; (void)n_in; (void)out_size; (void)ws_size;
    const float* logits  = (const float*)d_in[0];   // [64][8000][4] f32
    const float* targets = (const float*)d_in[1];   // [64][8000][4] f32
    const float* vmask   = (const float*)d_in[2];   // [64][8000]    f32

    float* costPart  = (float*)d_ws;                             // S*64*16 floats
    float* denomPart = costPart + (size_t)S_SLICES * B_DIM * 16; // S*64 floats

    dim3 grid(B_DIM / 2, S_SLICES);   // 32 batch pairs x 50 T-slices
    dim3 block(32);                   // one wave32 per block
    pit_cost_wmma<<<grid, block, 0, stream>>>(logits, targets, vmask,
                                              costPart, denomPart);
    pit_finalize<<<1, B_DIM, 0, stream>>>(costPart, denomPart, (float*)d_out);
}